// TokenSparse_73581379715507
// MI455X (gfx1250) — compile-verified
//
#include <hip/hip_runtime.h>
#include <hip/hip_bf16.h>
#include <math.h>

// Problem constants (from reference): B=32, L=4096, C=512,
// NUM_KEEP = ceil(4096*0.6) = 2458, NUM_NON = 1638.
#define B_        32
#define L_        4096
#define C_        512
#define NUM_KEEP_ 2458
#define NUM_NON_  (L_ - NUM_KEEP_)
#define ROWS_TOTAL (B_ * NUM_KEEP_)
#define MAX_CHUNKS 32

// ---------------------------------------------------------------------------
// CDNA5 async global<->LDS path (ASYNCcnt-tracked data mover).
// Builtins confirmed present: (AS1 v4i* global, AS3 v4i* lds, imm, imm).
// ---------------------------------------------------------------------------
#if __has_builtin(__builtin_amdgcn_global_load_async_to_lds_b128) && \
    __has_builtin(__builtin_amdgcn_global_store_async_from_lds_b128)
#define USE_ASYNC_LDS 1
#else
#define USE_ASYNC_LDS 0
#endif

typedef __attribute__((ext_vector_type(4))) int v4i;
typedef __attribute__((address_space(1))) v4i* global_v4i_ptr;
typedef __attribute__((address_space(3))) v4i* lds_v4i_ptr;
typedef unsigned long long u64;
typedef unsigned int u32;

#if USE_ASYNC_LDS
__device__ __forceinline__ void wait_asynccnt0() {
#if __has_builtin(__builtin_amdgcn_s_wait_asynccnt)
  __builtin_amdgcn_s_wait_asynccnt(0);
#else
  asm volatile("s_wait_asynccnt 0" ::: "memory");
#endif
}
#endif

// Monotonically DECREASING orderable mapping of a float (so ascending u32
// order == descending float order). NaNs absent in this problem.
__device__ __forceinline__ u32 key_desc(float f) {
  u32 u = __float_as_uint(f);
  u32 asc = (u & 0x80000000u) ? ~u : (u | 0x80000000u);  // monotone increasing
  return ~asc;                                            // monotone decreasing
}
__device__ __forceinline__ float key_desc_inv(u32 k) {
  u32 asc = ~k;
  u32 u = (asc & 0x80000000u) ? (asc ^ 0x80000000u) : ~asc;
  return __uint_as_float(u);
}

// ---------------------------------------------------------------------------
// Kernel 1: per-batch score = ax+ay, packed-u64 LDS bitonic argsort
// (ascending on (key_desc(score)<<32)|idx == descending score, stable),
// then tail softmax weights. One 1024-thread workgroup per batch.
// LDS: 32KB packed keys + 4KB reduction = 36KB (WGP has 320KB).
// ---------------------------------------------------------------------------
__global__ __launch_bounds__(1024) void score_sort_kernel(
    const float* __restrict__ ax, const float* __restrict__ ay,
    int* __restrict__ order, float* __restrict__ wts)
{
  __shared__ u64   a[L_];
  __shared__ float red[1024];

  const int b   = blockIdx.x;
  const int tid = threadIdx.x;

  for (int l = tid; l < L_; l += 1024) {
    const float sc = ax[b * L_ + l] + ay[b * L_ + l];
    a[l] = ((u64)key_desc(sc) << 32) | (u32)l;
  }
  __syncthreads();

  // Bitonic sort, ascending on packed u64 (branchless compare-exchange).
  for (int k = 2; k <= L_; k <<= 1) {
    for (int j = k >> 1; j > 0; j >>= 1) {
      #pragma unroll
      for (int m = 0; m < L_ / 1024; ++m) {
        const int t   = tid + m * 1024;
        const int ixj = t ^ j;
        if (ixj > t) {
          const u64 x = a[t], y = a[ixj];
          const u64 lo = (x < y) ? x : y;
          const u64 hi = (x < y) ? y : x;
          const bool up = ((t & k) == 0);
          a[t]   = up ? lo : hi;
          a[ixj] = up ? hi : lo;
        }
      }
      __syncthreads();
    }
  }

  for (int l = tid; l < L_; l += 1024) order[b * L_ + l] = (int)(u32)a[l];

  // Softmax over the sorted tail; max of tail is its first element (desc).
  const float m = key_desc_inv((u32)(a[NUM_KEEP_] >> 32));
  float part = 0.0f;
  for (int l = NUM_KEEP_ + tid; l < L_; l += 1024)
    part += expf(key_desc_inv((u32)(a[l] >> 32)) - m);
  red[tid] = part;
  __syncthreads();
  for (int off = 512; off > 0; off >>= 1) {
    if (tid < off) red[tid] += red[tid + off];
    __syncthreads();
  }
  const float inv = 1.0f / red[0];
  for (int l = NUM_KEEP_ + tid; l < L_; l += 1024)
    wts[b * NUM_NON_ + (l - NUM_KEEP_)] =
        expf(key_desc_inv((u32)(a[l] >> 32)) - m) * inv;
}

// ---------------------------------------------------------------------------
// Kernel 2a: partial weighted sums over a chunk of the non-kept rows.
// grid = (B, nchunk); block = 256; each thread owns 2 adjacent columns.
// partial[b][chunk][c] goes to scratch; ~8K waves keep HBM busy.
// ---------------------------------------------------------------------------
__global__ __launch_bounds__(256) void extra_partial_kernel(
    const float* __restrict__ tokens, const int* __restrict__ order,
    const float* __restrict__ wts, float* __restrict__ partial,
    int chunk_len)
{
  const int b  = blockIdx.x;
  const int ci = blockIdx.y;
  const int tid = threadIdx.x;
  const int c0  = tid * 2;

  const int j0 = ci * chunk_len;
  int j1 = j0 + chunk_len; if (j1 > NUM_NON_) j1 = NUM_NON_;

  const int*   ord = order + b * L_ + NUM_KEEP_;
  const float* w   = wts + b * NUM_NON_;
  const float* tb  = tokens + (size_t)b * L_ * C_;

  float acc0 = 0.0f, acc1 = 0.0f;
  #pragma unroll 4
  for (int j = j0; j < j1; ++j) {
    const int   idx = ord[j];
    const float wj  = w[j];
    const float2 v  = *(const float2*)(tb + (size_t)idx * C_ + c0);
    acc0 = fmaf(wj, v.x, acc0);
    acc1 = fmaf(wj, v.y, acc1);
  }
  float* p = partial + ((size_t)b * gridDim.y + ci) * C_;
  p[c0]     = acc0;
  p[c0 + 1] = acc1;
}

// Kernel 2b: reduce partials -> extra_token. grid = B, block = 256.
__global__ __launch_bounds__(256) void extra_reduce_kernel(
    const float* __restrict__ partial, float* __restrict__ out_extra,
    int nchunk)
{
  const int b   = blockIdx.x;
  const int c0  = threadIdx.x * 2;
  float acc0 = 0.0f, acc1 = 0.0f;
  for (int ci = 0; ci < nchunk; ++ci) {
    const float* p = partial + ((size_t)b * nchunk + ci) * C_;
    acc0 += p[c0];
    acc1 += p[c0 + 1];
  }
  out_extra[b * C_ + c0]     = acc0;
  out_extra[b * C_ + c0 + 1] = acc1;
}

// ---------------------------------------------------------------------------
// Kernel 3: gather the kept rows (2KB each) in sorted order.
// One wave32 per row; 8 waves (256 threads) per block. Rows are staged
// through LDS with ASYNCcnt-tracked async B128 copies (lane = 16B, 4 chunks
// of 512B), s_wait_asynccnt 0, then async-stores from LDS. s_endpgm's
// implicit wait-idle covers completion of the final stores.
// ---------------------------------------------------------------------------
__global__ __launch_bounds__(256) void gather_kernel(
    const float* __restrict__ tokens, const int* __restrict__ order,
    float* __restrict__ out_sel)
{
  __shared__ __align__(16) float buf[8 * C_];   // 16 KB: 8 waves x 512 floats
  const int wave = threadIdx.x >> 5;
  const int lane = threadIdx.x & 31;
  const int row  = blockIdx.x * 8 + wave;
  if (row >= ROWS_TOTAL) return;

  const int b = row / NUM_KEEP_;
  const int k = row - b * NUM_KEEP_;
  const int idx = order[b * L_ + k];            // wave-uniform

  const float* src = tokens + ((size_t)b * L_ + idx) * C_;
  float*       dst = out_sel + (size_t)row * C_;

#if USE_ASYNC_LDS
  float* lbuf = buf + wave * C_;
  #pragma unroll
  for (int i = 0; i < 4; ++i) {
    const int off = i * 128 + lane * 4;         // 16B per lane per chunk
    __builtin_amdgcn_global_load_async_to_lds_b128(
        (global_v4i_ptr)(src + off), (lds_v4i_ptr)(lbuf + off), 0, 0);
  }
  wait_asynccnt0();                             // LDS writes visible to stores
  #pragma unroll
  for (int i = 0; i < 4; ++i) {
    const int off = i * 128 + lane * 4;
    __builtin_amdgcn_global_store_async_from_lds_b128(
        (global_v4i_ptr)(dst + off), (lds_v4i_ptr)(lbuf + off), 0, 0);
  }
#else
  (void)buf;
  #pragma unroll
  for (int i = 0; i < 4; ++i) {
    const int off = i * 128 + lane * 4;
    *(float4*)(dst + off) = *(const float4*)(src + off);
  }
#endif
}

// ---------------------------------------------------------------------------
// Launch. Inputs: tokens (B,L,C) f32, attention_x (B,L), attention_y (B,L).
// Output: select_tokens (B,NUM_KEEP,C) then extra (B,1,C), flat f32.
// Scratch layout: order (B*L i32) | wts (B*NUM_NON f32) | partials
// (B*nchunk*C f32, nchunk adapted to ws_size, <=32). All scratch rewritten
// before being read on every call => deterministic.
// ---------------------------------------------------------------------------
extern "C" void kernel_launch(void* const* d_in, const int* in_sizes, int n_in,
                              void* d_out, int out_size, void* d_ws, size_t ws_size,
                              hipStream_t stream) {
  const float* tokens = (const float*)d_in[0];
  const float* ax     = (const float*)d_in[1];
  const float* ay     = (const float*)d_in[2];

  float* out       = (float*)d_out;
  float* out_sel   = out;
  float* out_extra = out + (size_t)B_ * NUM_KEEP_ * C_;

  const size_t order_bytes = (size_t)B_ * L_ * sizeof(int);
  const size_t wts_bytes   = (size_t)B_ * NUM_NON_ * sizeof(float);
  int*   order    = (int*)d_ws;
  float* wts      = (float*)((char*)d_ws + order_bytes);
  float* partials = (float*)((char*)d_ws + order_bytes + wts_bytes);

  // Adapt chunk count to available scratch (deterministic for fixed ws_size).
  size_t avail = (ws_size > order_bytes + wts_bytes)
                     ? (ws_size - order_bytes - wts_bytes) : 0;
  int nchunk = (int)(avail / ((size_t)B_ * C_ * sizeof(float)));
  if (nchunk > MAX_CHUNKS) nchunk = MAX_CHUNKS;
  if (nchunk < 1) nchunk = 1;
  const int chunk_len = (NUM_NON_ + nchunk - 1) / nchunk;

  hipLaunchKernelGGL(score_sort_kernel, dim3(B_), dim3(1024), 0, stream,
                     ax, ay, order, wts);
  hipLaunchKernelGGL(extra_partial_kernel, dim3(B_, nchunk), dim3(256), 0, stream,
                     tokens, order, wts, partials, chunk_len);
  hipLaunchKernelGGL(extra_reduce_kernel, dim3(B_), dim3(256), 0, stream,
                     partials, out_extra, nchunk);
  const int nblocks = (ROWS_TOTAL + 7) / 8;
  hipLaunchKernelGGL(gather_kernel, dim3(nblocks), dim3(256), 0, stream,
                     tokens, order, out_sel);
}